// LaplacianBuilder_18957985644862
// MI455X (gfx1250) — compile-verified
//
#include <hip/hip_runtime.h>

// LaplacianBuilder for MI455X (gfx1250, wave32).
// Output is 8192x8192 f32 (256 MB): bandwidth floor ~11us @ 23.3 TB/s HBM.
// Batched 4x4 A^T*B products run on V_WMMA_F32_16X16X4_F32 (4 edges/wave).

#define DD 4

typedef __attribute__((ext_vector_type(2))) float v2f;
typedef __attribute__((ext_vector_type(4))) float v4f;
typedef __attribute__((ext_vector_type(8))) float v8f;

// ---------------- zero-fill 256 MB output with NT 128-bit stores ----------------
__global__ void zero_out_kernel(v4f* __restrict__ out, long nvec) {
  long i = (long)blockIdx.x * blockDim.x + threadIdx.x;
  long stride = (long)gridDim.x * blockDim.x;
  v4f z = {0.f, 0.f, 0.f, 0.f};
  for (; i < nvec; i += stride)
    __builtin_nontemporal_store(z, out + i);
}

__global__ void zero_f32_kernel(float* __restrict__ p, int n) {
  int i = blockIdx.x * blockDim.x + threadIdx.x;
  if (i < n) p[i] = 0.f;
}

// Extract this lane's 4-element block column from the WMMA accumulator.
// r-half selected by (g&1); constant indices keep it as cndmasks, not a
// dynamic-index select tree.
__device__ __forceinline__ void extract_col(const v8f& c, int g, float x[4]) {
  if (g & 1) {
    x[0] = c[4]; x[1] = c[5]; x[2] = c[6]; x[3] = c[7];
  } else {
    x[0] = c[0]; x[1] = c[1]; x[2] = c[2]; x[3] = c[3];
  }
}

// ---------------- K1: diag_contrib = M^T M per edge, atomic segment-sum --------
// 4 edges per wave. A(16x4): lane s<16 => row m=s (edge g=s/4, col i=s%4),
// VGPR v + lane-half h => K = h*2+v. B(4x16) has the identical per-lane layout
// (lane = N, {v,h} = K), and for the Gram product the B element equals the A
// element, so one 2-VGPR operand feeds both WMMA sources.
__global__ void gram_accum_wmma(const float* __restrict__ maps,
                                const int* __restrict__ row_all,
                                float* __restrict__ diag,   // N*16 accumulator
                                int twoE) {
  int tid  = blockIdx.x * blockDim.x + threadIdx.x;
  int wave = tid >> 5;
  int lane = tid & 31;
  int e0 = wave << 2;            // first of 4 edges for this wave
  if (e0 >= twoE) return;        // wave-uniform: EXEC stays all-ones for WMMA
  int s = lane & 15, half = lane >> 4;
  int g = s >> 2, ij = s & 3;    // block (edge-in-wave), column index
  int e = e0 + g;
  const float* mp = maps + e * 16 + half * 8 + ij;   // M[k0][ij], M[k0+1][ij]
  v2f a;
  a.x = mp[0];
  a.y = mp[4];
  v8f c = {};
  c = __builtin_amdgcn_wmma_f32_16x16x4_f32(false, a, false, a, (short)0, c,
                                            false, false);
  // Diagonal block g lives in rows 4g..4g+3: half==g>>1 lanes, VGPRs (g&1)*4..+3
  if (half == (g >> 1)) {
    float x[4];
    extract_col(c, g, x);
    int node = row_all[e];
    float* dst = diag + node * 16 + ij;  // column ij of node's 4x4 Gram sum
    atomicAdd(dst + 0,  x[0]);
    atomicAdd(dst + 4,  x[1]);
    atomicAdd(dst + 8,  x[2]);
    atomicAdd(dst + 12, x[3]);
  }
}

// ---------------- K2: maps_triu[e] = M_fwd^T M_rev via WMMA -------------------
__global__ void triu_wmma(const float* __restrict__ maps,
                          float* __restrict__ triu,   // E*16
                          int E) {
  int tid  = blockIdx.x * blockDim.x + threadIdx.x;
  int wave = tid >> 5;
  int lane = tid & 31;
  int e0 = wave << 2;
  if (e0 >= E) return;
  int s = lane & 15, half = lane >> 4;
  int g = s >> 2, ij = s & 3;
  int e = e0 + g;
  const float* ma = maps + e * 16 + half * 8 + ij;             // forward map
  const float* mb = maps + (e + E) * 16 + half * 8 + ij;       // reverse map
  v2f a, b;
  a.x = ma[0]; a.y = ma[4];
  b.x = mb[0]; b.y = mb[4];
  v8f c = {};
  c = __builtin_amdgcn_wmma_f32_16x16x4_f32(false, a, false, b, (short)0, c,
                                            false, false);
  if (half == (g >> 1)) {
    float x[4];
    extract_col(c, g, x);
    float* dst = triu + e * 16 + ij;
    dst[0]  = x[0];
    dst[4]  = x[1];
    dst[8]  = x[2];
    dst[12] = x[3];
  }
}

// ---------------- K3: per-node Jacobi eigensolve -> inv-sqrt, diag blocks -----
#define JROT(p, q)                                                             \
  {                                                                            \
    float apq = a[p][q];                                                       \
    if (fabsf(apq) > 1e-12f) {                                                 \
      float tau = (a[q][q] - a[p][p]) / (2.0f * apq);                          \
      float t = (tau >= 0.f ? 1.f : -1.f) /                                    \
                (fabsf(tau) + sqrtf(1.f + tau * tau));                         \
      float cc = 1.0f / sqrtf(1.f + t * t);                                    \
      float ss = t * cc;                                                       \
      _Pragma("unroll") for (int k = 0; k < 4; ++k) {                          \
        float akp = a[k][p], akq = a[k][q];                                    \
        a[k][p] = cc * akp - ss * akq;                                         \
        a[k][q] = ss * akp + cc * akq;                                         \
      }                                                                        \
      _Pragma("unroll") for (int k = 0; k < 4; ++k) {                          \
        float apk = a[p][k], aqk = a[q][k];                                    \
        a[p][k] = cc * apk - ss * aqk;                                         \
        a[q][k] = ss * apk + cc * aqk;                                         \
      }                                                                        \
      _Pragma("unroll") for (int k = 0; k < 4; ++k) {                          \
        float vkp = v[k][p], vkq = v[k][q];                                    \
        v[k][p] = cc * vkp - ss * vkq;                                         \
        v[k][q] = ss * vkp + cc * vkq;                                         \
      }                                                                        \
    }                                                                          \
  }

__global__ void invsqrt_diag_kernel(const float* __restrict__ diag,
                                    float* __restrict__ invs,
                                    float* __restrict__ out,
                                    int N, int n4) {
  int n = blockIdx.x * blockDim.x + threadIdx.x;
  if (n >= N) return;
  float D0[4][4], a[4][4], v[4][4];
#pragma unroll
  for (int i = 0; i < 4; ++i) {
    v4f r = *(const v4f*)(diag + n * 16 + i * 4);
#pragma unroll
    for (int j = 0; j < 4; ++j) {
      D0[i][j] = r[j];
      a[i][j]  = r[j] + (i == j ? 1.0f : 0.0f);  // to_inv = maps_diag + I
      v[i][j]  = (i == j) ? 1.0f : 0.0f;
    }
  }
  for (int sw = 0; sw < 6; ++sw) {
    JROT(0, 1) JROT(0, 2) JROT(0, 3) JROT(1, 2) JROT(1, 3) JROT(2, 3)
  }
  float w[4];
#pragma unroll
  for (int i = 0; i < 4; ++i)
    w[i] = rsqrtf(fmaxf(a[i][i], 1e-8f));   // clip(evals,1e-8)^(-1/2)
  float S[4][4];
#pragma unroll
  for (int i = 0; i < 4; ++i) {
    v4f r;
#pragma unroll
    for (int j = 0; j < 4; ++j) {
      float acc = 0.f;
#pragma unroll
      for (int k = 0; k < 4; ++k) acc += v[i][k] * w[k] * v[j][k];
      S[i][j] = acc;
      r[j] = acc;
    }
    *(v4f*)(invs + n * 16 + i * 4) = r;
  }
  // diag block: clip(S * D0 * S), written at (n,n)
  float U[4][4];
#pragma unroll
  for (int i = 0; i < 4; ++i)
#pragma unroll
    for (int j = 0; j < 4; ++j) {
      float acc = 0.f;
#pragma unroll
      for (int k = 0; k < 4; ++k) acc += S[i][k] * D0[k][j];
      U[i][j] = acc;
    }
#pragma unroll
  for (int i = 0; i < 4; ++i) {
    v4f r;
#pragma unroll
    for (int j = 0; j < 4; ++j) {
      float acc = 0.f;
#pragma unroll
      for (int k = 0; k < 4; ++k) acc += U[i][k] * S[k][j];
      r[j] = fminf(fmaxf(acc, -1.0f), 1.0f);
    }
    *(v4f*)(out + (long)(n * 4 + i) * n4 + n * 4) = r;   // 16B-aligned b128
  }
}

// ---------------- K4: T = S[row]*P*S[col], clip, scatter off-diag blocks ------
__global__ void sandwich_scatter_kernel(const int* __restrict__ eidx,
                                        const float* __restrict__ invs,
                                        const float* __restrict__ triu,
                                        float* __restrict__ out,
                                        int E, int twoE, int n4) {
  int e = blockIdx.x * blockDim.x + threadIdx.x;
  if (e >= E) return;
  int row = eidx[e];          // edge_index[0][e]
  int col = eidx[twoE + e];   // edge_index[1][e]
  float Sr[4][4], Sc[4][4], P[4][4];
#pragma unroll
  for (int i = 0; i < 4; ++i) {
    v4f ra = *(const v4f*)(invs + row * 16 + i * 4);
    v4f rb = *(const v4f*)(invs + col * 16 + i * 4);
    v4f rp = *(const v4f*)(triu + e * 16 + i * 4);
#pragma unroll
    for (int j = 0; j < 4; ++j) {
      Sr[i][j] = ra[j];
      Sc[i][j] = rb[j];
      P[i][j]  = rp[j];
    }
  }
  float U[4][4], T[4][4];
#pragma unroll
  for (int i = 0; i < 4; ++i)
#pragma unroll
    for (int j = 0; j < 4; ++j) {
      float acc = 0.f;
#pragma unroll
      for (int k = 0; k < 4; ++k) acc += Sr[i][k] * P[k][j];
      U[i][j] = acc;
    }
#pragma unroll
  for (int i = 0; i < 4; ++i)
#pragma unroll
    for (int j = 0; j < 4; ++j) {
      float acc = 0.f;
#pragma unroll
      for (int k = 0; k < 4; ++k) acc += U[i][k] * Sc[k][j];
      T[i][j] = fminf(fmaxf(acc, -1.0f), 1.0f);
    }
  // L[row,col] = -T ; L[col,row] = -T^T  (no collisions: edges unique, no loops)
  // Each block row is a 16B-aligned contiguous run -> b128 stores.
#pragma unroll
  for (int i = 0; i < 4; ++i) {
    v4f r1, r2;
#pragma unroll
    for (int j = 0; j < 4; ++j) {
      r1[j] = -T[i][j];
      r2[j] = -T[j][i];
    }
    *(v4f*)(out + (long)(row * 4 + i) * n4 + col * 4) = r1;
    *(v4f*)(out + (long)(col * 4 + i) * n4 + row * 4) = r2;
  }
}

extern "C" void kernel_launch(void* const* d_in, const int* in_sizes, int n_in,
                              void* d_out, int out_size, void* d_ws, size_t ws_size,
                              hipStream_t stream) {
  const float* maps = (const float*)d_in[0];
  const int*   eidx = (const int*)d_in[1];
  int twoE = in_sizes[0] / 16;   // 65536
  int E    = twoE / 2;           // 32768
  int N    = 2048;               // fixed by reference (out = (N*4)^2)
  int n4   = N * DD;             // 8192
  float* out = (float*)d_out;

  // workspace: [diag N*16][invs N*16][triu E*16] floats = ~2.4 MB
  float* diag = (float*)d_ws;
  float* invs = diag + (size_t)N * 16;
  float* triu = invs + (size_t)N * 16;

  // 1) zero 256 MB output (bandwidth-dominant step, NT b128 stores)
  long nvec = (long)out_size / 4;
  zero_out_kernel<<<4096, 256, 0, stream>>>((v4f*)out, nvec);

  // 2) zero the per-node Gram accumulator
  zero_f32_kernel<<<(N * 16 + 255) / 256, 256, 0, stream>>>(diag, N * 16);

  // 3) WMMA Gram products + atomic segment-sum over all 2E maps
  {
    int waves = (twoE + 3) / 4;
    int blocks = (waves * 32 + 255) / 256;
    gram_accum_wmma<<<blocks, 256, 0, stream>>>(maps, eidx, diag, twoE);
  }

  // 4) WMMA triu products for the E undirected edges
  {
    int waves = (E + 3) / 4;
    int blocks = (waves * 32 + 255) / 256;
    triu_wmma<<<blocks, 256, 0, stream>>>(maps, triu, E);
  }

  // 5) per-node Jacobi eigensolve -> inv-sqrt + diagonal blocks
  invsqrt_diag_kernel<<<(N + 255) / 256, 256, 0, stream>>>(diag, invs, out, N, n4);

  // 6) sandwich + scatter off-diagonal blocks
  sandwich_scatter_kernel<<<(E + 255) / 256, 256, 0, stream>>>(eidx, invs, triu,
                                                               out, E, twoE, n4);
}